// PyTorchFFTConv_73675868996060
// MI455X (gfx1250) — compile-verified
//
#include <hip/hip_runtime.h>

typedef __attribute__((ext_vector_type(2))) float v2f;
typedef __attribute__((ext_vector_type(4))) float v4f;
typedef __attribute__((ext_vector_type(8))) float v8f;
typedef __attribute__((ext_vector_type(4))) int   v4i;

#define N_FFT 16384
#define L_SIG 8192
#define NT    512
#define NWAVE (NT / 32)

#if __has_builtin(__builtin_amdgcn_global_load_async_to_lds_b128)
#define HAVE_ASYNC_LDS 1
#else
#define HAVE_ASYNC_LDS 0
#endif

#if HAVE_ASYNC_LDS
// Builtin signature (from hipcc diagnostic): params are pointers to int4.
typedef __attribute__((address_space(1))) v4i GAS_v4i;  // global int4
typedef __attribute__((address_space(3))) v4i LAS_v4i;  // LDS int4

__device__ __forceinline__ void async_g2l_b128(const float* g, float* l) {
  __builtin_amdgcn_global_load_async_to_lds_b128((GAS_v4i*)g, (LAS_v4i*)l, 0,
                                                 0);
}
__device__ __forceinline__ void wait_async() {
#if __has_builtin(__builtin_amdgcn_s_wait_asynccnt)
  __builtin_amdgcn_s_wait_asynccnt(0);
#else
  asm volatile("s_wait_asynccnt 0" ::: "memory");
#endif
}
#endif

__device__ __forceinline__ v8f wmma4(v2f a, v2f b, v8f c) {
  // D(16x16,f32) = A(16x4,f32) * B(4x16,f32) + C
  return __builtin_amdgcn_wmma_f32_16x16x4_f32(false, a, false, b, (short)0, c,
                                               false, false);
}

// In-place radix-16 stage on LDS arrays. fwd: DIF (post-twiddle, exp(-i..)).
// !fwd: DIT (pre-twiddle, exp(+i..)) with conjugate DFT matrix (sgn=-1).
template <int L>
__device__ void radix16_stage(float* re, float* im, bool fwd, float sgn,
                              int tid, const float* aC, const float* aS) {
  constexpr int J = L / 16;
  const int lane = tid & 31;
  const int wave = tid >> 5;
  const int hi   = lane >> 4;   // which half of the wave
  const int col  = lane & 15;   // butterfly column within group / M row for A
  const float twopi = 6.28318530717958647692f;

  // N/16 = 1024 butterflies, 16 per group -> 64 groups over 16 waves
  for (int gi = 0; gi < (N_FFT / 16 / 16) / NWAVE; ++gi) {
    const int g    = wave * ((N_FFT / 16 / 16) / NWAVE) + gi;
    const int bf   = g * 16 + col;
    const int j    = bf & (J - 1);
    const int b    = bf / J;
    const int base = b * L + j;

    // Inverse pre-twiddle generator: w = exp(+i*2pi*j*t/L), t steps +1,+3,...
    float wr = 1.f, wi = 0.f, c1 = 1.f, s1 = 0.f, c3 = 1.f, s3 = 0.f;
    if (!fwd) {
      float th1 = twopi * (float)j / (float)L;
      __sincosf(th1, &s1, &c1);                       // w1 = exp(+i*2pi*j/L)
      float th0 = twopi * (float)(2 * hi * j) / (float)L;
      float s0, c0;
      __sincosf(th0, &s0, &c0);                       // start at t = 2*hi
      wr = c0; wi = s0;
      float c2 = c1 * c1 - s1 * s1, s2 = 2.f * c1 * s1;
      c3 = c2 * c1 - s2 * s1;                          // w3 = w1^3
      s3 = c2 * s1 + s2 * c1;
    }

    // Gather B fragments: this lane supplies rows t = 4c + 2*hi + r, col 'col'
    float xr[8], xi[8];
#pragma unroll
    for (int c = 0; c < 4; ++c) {
#pragma unroll
      for (int r = 0; r < 2; ++r) {
        const int t = 4 * c + 2 * hi + r;
        float vr = re[base + t * J];
        float vi = im[base + t * J];
        if (!fwd) {
          float tr = vr * wr - vi * wi;
          float ti = vr * wi + vi * wr;
          vr = tr; vi = ti;
          if (r == 0) {  // advance t by 1
            float nwr = wr * c1 - wi * s1;
            wi = wr * s1 + wi * c1; wr = nwr;
          } else {       // advance t by 3 (to next chunk)
            float nwr = wr * c3 - wi * s3;
            wi = wr * s3 + wi * c3; wr = nwr;
          }
        }
        xr[2 * c + r] = vr;
        xi[2 * c + r] = vi;
      }
    }

    // Yr = C*Xr + sgn*S*Xi ; Yi = C*Xi - sgn*S*Xr  (W = cos - i*sin, conj inv)
    v8f yr = {0.f, 0.f, 0.f, 0.f, 0.f, 0.f, 0.f, 0.f};
    v8f yi = {0.f, 0.f, 0.f, 0.f, 0.f, 0.f, 0.f, 0.f};
#pragma unroll
    for (int c = 0; c < 4; ++c) {
      v2f aCc = {aC[2 * c], aC[2 * c + 1]};
      v2f aSp = {sgn * aS[2 * c], sgn * aS[2 * c + 1]};
      v2f aSn = {-aSp.x, -aSp.y};
      v2f br  = {xr[2 * c], xr[2 * c + 1]};
      v2f bi  = {xi[2 * c], xi[2 * c + 1]};
      yr = wmma4(aCc, br, yr);
      yr = wmma4(aSp, bi, yr);
      yi = wmma4(aCc, bi, yi);
      yi = wmma4(aSn, br, yi);
    }

    // Forward post-twiddle generator: w = exp(-i*2pi*j*m/L), m = 8*hi + v
    float cw = 1.f, sw = 0.f;
    if (fwd) {
      float th1 = -twopi * (float)j / (float)L;
      __sincosf(th1, &sw, &cw);                       // step exp(-i*2pi*j/L)
      float th0 = -twopi * (float)(8 * hi * j) / (float)L;
      float s0, c0;
      __sincosf(th0, &s0, &c0);                       // start at m = 8*hi
      wr = c0; wi = s0;
    }

    // Scatter D: lane holds rows m = v + 8*hi of column 'col'
#pragma unroll
    for (int v = 0; v < 8; ++v) {
      const int m = v + 8 * hi;
      float a = yr[v], bb = yi[v];
      float orv = a, oiv = bb;
      if (fwd) {
        orv = a * wr - bb * wi;
        oiv = a * wi + bb * wr;
        float nwr = wr * cw - wi * sw;
        wi = wr * sw + wi * cw; wr = nwr;
      }
      re[base + m * J] = orv;
      im[base + m * J] = oiv;
    }
  }
}

// Final (fwd) / first (inv) radix-4 stage, L=4, unit twiddles. Plain VALU.
__device__ void radix4_stage(float* re, float* im, float sgn, int tid) {
  for (int it = 0; it < (N_FFT / 4) / NT; ++it) {
    const int q = tid + it * NT;
    const int p = q * 4;
    float x0r = re[p + 0], x0i = im[p + 0];
    float x1r = re[p + 1], x1i = im[p + 1];
    float x2r = re[p + 2], x2i = im[p + 2];
    float x3r = re[p + 3], x3i = im[p + 3];
    float d0r = x0r + x2r, d0i = x0i + x2i;
    float d1r = x0r - x2r, d1i = x0i - x2i;
    float e0r = x1r + x3r, e0i = x1i + x3i;
    float e1r = x1r - x3r, e1i = x1i - x3i;
    re[p + 0] = d0r + e0r;           im[p + 0] = d0i + e0i;
    re[p + 2] = d0r - e0r;           im[p + 2] = d0i - e0i;
    re[p + 1] = d1r + sgn * e1i;     im[p + 1] = d1i - sgn * e1r;
    re[p + 3] = d1r - sgn * e1i;     im[p + 3] = d1i + sgn * e1r;
  }
}

__device__ void fft16384(float* re, float* im, bool fwd, int tid,
                         const float* aC, const float* aS) {
  const float sgn = fwd ? 1.f : -1.f;
  if (fwd) {  // DIF: natural in -> digit-reversed out
    radix16_stage<16384>(re, im, true, sgn, tid, aC, aS); __syncthreads();
    radix16_stage<1024 >(re, im, true, sgn, tid, aC, aS); __syncthreads();
    radix16_stage<64   >(re, im, true, sgn, tid, aC, aS); __syncthreads();
    radix4_stage(re, im, sgn, tid);                       __syncthreads();
  } else {    // DIT with conj twiddles: digit-reversed in -> natural out
    radix4_stage(re, im, sgn, tid);                       __syncthreads();
    radix16_stage<64   >(re, im, false, sgn, tid, aC, aS); __syncthreads();
    radix16_stage<1024 >(re, im, false, sgn, tid, aC, aS); __syncthreads();
    radix16_stage<16384>(re, im, false, sgn, tid, aC, aS); __syncthreads();
  }
}

__global__ __launch_bounds__(NT) void fftconv_wmma_kernel(
    const float* __restrict__ u, const float* __restrict__ k,
    float* __restrict__ y) {
  extern __shared__ float smem[];
  float* re = smem;
  float* im = smem + N_FFT;

  const int tid  = threadIdx.x;
  const int h    = blockIdx.x >> 2;   // 1024 heads
  const int pair = blockIdx.x & 3;    // 4 batch pairs
  const int b0   = 2 * pair, b1 = 2 * pair + 1;

  // Per-lane constant A fragments of the DFT-16 matrix (16x4 A layout:
  // lane<16 -> K = 4c+{0,1}, lane>=16 -> K = 4c+{2,3}; row M = lane&15).
  float aC[8], aS[8];
  {
    const int lane = tid & 31, hi = lane >> 4, m = lane & 15;
#pragma unroll
    for (int c = 0; c < 4; ++c) {
#pragma unroll
      for (int r = 0; r < 2; ++r) {
        const int t = 4 * c + 2 * hi + r;
        const float th =
            6.28318530717958647692f * (float)((m * t) & 15) * (1.0f / 16.0f);
        __sincosf(th, &aS[2 * c + r], &aC[2 * c + r]);
      }
    }
  }

  const v4f z4 = {0.f, 0.f, 0.f, 0.f};

  // ---- 1. FFT of k[h] (zero-padded, imag = 0) ----
  {
    const float* kh = k + (size_t)h * L_SIG;
#pragma unroll
    for (int it = 0; it < (L_SIG / 4) / NT; ++it) {  // 4 x b128 per thread
      const int n = 4 * (tid + it * NT);
#if HAVE_ASYNC_LDS
      async_g2l_b128(kh + n, re + n);
#else
      *(v4f*)(re + n) = *(const v4f*)(kh + n);
#endif
      *(v4f*)(re + L_SIG + n) = z4;  // zero pad upper half
    }
#pragma unroll
    for (int it = 0; it < (N_FFT / 4) / NT; ++it) {
      const int n = 4 * (tid + it * NT);
      *(v4f*)(im + n) = z4;
    }
#if HAVE_ASYNC_LDS
    wait_async();
#endif
    __syncthreads();
  }
  fft16384(re, im, true, tid, aC, aS);

  // Cache K spectrum (digit-reversed order) in registers: 32 complex/lane.
  // Stride-NT indexing keeps the b32 DS reads bank-conflict free.
  float krg[32], kig[32];
#pragma unroll
  for (int s = 0; s < 32; ++s) {
    krg[s] = re[tid + s * NT];
    kig[s] = im[tid + s * NT];
  }
  __syncthreads();

  // ---- 2. Packed FFT of z = u[b0] + i*u[b1] (zero-padded) ----
  {
    const float* u0 = u + ((size_t)b0 * 1024 + h) * L_SIG;
    const float* u1 = u + ((size_t)b1 * 1024 + h) * L_SIG;
#pragma unroll
    for (int it = 0; it < (L_SIG / 4) / NT; ++it) {
      const int n = 4 * (tid + it * NT);
#if HAVE_ASYNC_LDS
      async_g2l_b128(u0 + n, re + n);
      async_g2l_b128(u1 + n, im + n);
#else
      *(v4f*)(re + n) = *(const v4f*)(u0 + n);
      *(v4f*)(im + n) = *(const v4f*)(u1 + n);
#endif
      *(v4f*)(re + L_SIG + n) = z4;
      *(v4f*)(im + L_SIG + n) = z4;
    }
#if HAVE_ASYNC_LDS
    wait_async();
#endif
    __syncthreads();
  }
  fft16384(re, im, true, tid, aC, aS);

  // ---- 3. Pointwise Z*K/N (valid in matching digit-reversed domain) ----
  const float invN = 1.0f / (float)N_FFT;
#pragma unroll
  for (int s = 0; s < 32; ++s) {
    const int n = tid + s * NT;
    const float zr = re[n], zi = im[n];
    re[n] = (zr * krg[s] - zi * kig[s]) * invN;
    im[n] = (zr * kig[s] + zi * krg[s]) * invN;
  }
  __syncthreads();

  // ---- 4. Inverse FFT; Re -> y[b0], Im -> y[b1] (both convs are real) ----
  fft16384(re, im, false, tid, aC, aS);

  float* y0 = y + ((size_t)b0 * 1024 + h) * L_SIG;
  float* y1 = y + ((size_t)b1 * 1024 + h) * L_SIG;
#pragma unroll
  for (int it = 0; it < (L_SIG / 4) / NT; ++it) {
    const int n = 4 * (tid + it * NT);
    *(v4f*)(y0 + n) = *(const v4f*)(re + n);
    *(v4f*)(y1 + n) = *(const v4f*)(im + n);
  }
}

extern "C" void kernel_launch(void* const* d_in, const int* in_sizes, int n_in,
                              void* d_out, int out_size, void* d_ws,
                              size_t ws_size, hipStream_t stream) {
  (void)in_sizes; (void)n_in; (void)out_size; (void)d_ws; (void)ws_size;
  const float* u = (const float*)d_in[0];
  const float* k = (const float*)d_in[1];
  float* y = (float*)d_out;

  const size_t lds_bytes = 2 * N_FFT * sizeof(float);  // 128 KB
  (void)hipFuncSetAttribute(reinterpret_cast<const void*>(fftconv_wmma_kernel),
                            hipFuncAttributeMaxDynamicSharedMemorySize,
                            (int)lds_bytes);

  dim3 grid(1024 * 4);  // heads * batch-pairs
  dim3 block(NT);
  fftconv_wmma_kernel<<<grid, block, lds_bytes, stream>>>(u, k, y);
}